// ALS_net_14602888807029
// MI455X (gfx1250) — compile-verified
//
#include <hip/hip_runtime.h>

typedef __attribute__((ext_vector_type(2))) float v2f;
typedef __attribute__((ext_vector_type(8))) float v8f;

#define K_DIM     64
#define GOODS_N   100000

// One wave32 computes 16 pair-dot-products as the diagonal of a 16x16 WMMA
// accumulation: D = sum_kk A(16x4) x B(4x16), A = gathered user rows,
// B = gathered goods columns. Purely HBM-bound (~295 MB total), so the 16x
// off-diagonal compute waste is free and keeps full f32 precision.
//
// Cache policy: user rows (256 MB, no reuse) + indices + output are marked
// non-temporal so the 192 MB L2 stays dedicated to the goods matrix
// (25.6 MB, ~10x reuse), which is loaded with default (RT) policy.
__global__ __launch_bounds__(128)
void pair_dot_wmma_f32(const int*   __restrict__ loc,   // (N_PAIRS, 2) int32
                       const float* __restrict__ U,     // (USER_NUM, 64)
                       const float* __restrict__ G,     // (64, GOODS_N)
                       float*       __restrict__ out,   // (N_PAIRS)
                       int n_pairs)
{
    const int lane = threadIdx.x & 31;
    const int wave = threadIdx.x >> 5;
    const int tile = blockIdx.x * 4 + wave;
    const int base = tile * 16;                 // first pair of this tile
    if (base >= n_pairs) return;                // whole-wave uniform exit

    // Lane L serves pair (L & 15); clamp keeps loads in-bounds for a ragged
    // tail without creating divergent EXEC before the WMMAs.
    int p = base + (lane & 15);
    p = p < n_pairs ? p : n_pairs - 1;

    // Index stream: read once, non-temporal.
    const int urow = __builtin_nontemporal_load(loc + 2 * p + 0);
    const int gcol = __builtin_nontemporal_load(loc + 2 * p + 1);

    // ISA 16x4 f32 A layout: lanes 0-15 -> K={kk,kk+1}, lanes 16-31 -> K={kk+2,kk+3}
    const int koff  = (lane >= 16) ? 2 : 0;
    const int ubase = urow * K_DIM + koff;      // fits in int (1e6*64)

    v8f c = {};
    #pragma unroll
    for (int kk = 0; kk < K_DIM; kk += 4) {
        // A: one float2 per lane per step (coalesced within each user row),
        // non-temporal: streamed, never reused.
        v2f a = __builtin_nontemporal_load((const v2f*)(U + ubase + kk));
        // B (4x16, mirrored half-wave K split): two 4B loads; default RT policy
        // keeps the 25.6 MB goods matrix resident in the 192 MB L2.
        v2f b;
        b.x = G[(kk + koff + 0) * GOODS_N + gcol];
        b.y = G[(kk + koff + 1) * GOODS_N + gcol];
        // D = A*B + C  (f32, round-to-nearest-even, matches reference precision)
        c = __builtin_amdgcn_wmma_f32_16x16x4_f32(
                /*neg_a=*/false, a, /*neg_b=*/false, b,
                /*c_mod=*/(short)0, c, /*reuse_a=*/false, /*reuse_b=*/false);
    }

    // Diagonal extraction from 16x16 f32 C/D layout:
    //   D[i][i], i<8  -> VGPR i,   lane i
    //   D[i][i], i>=8 -> VGPR i-8, lane 16+i  (i.e. lane 24+v, v=i-8)
    float* o = out + base;
    #pragma unroll
    for (int v = 0; v < 8; ++v) {
        if (lane == v && (base + v) < n_pairs)
            __builtin_nontemporal_store(c[v], o + v);
        if (lane == 24 + v && (base + 8 + v) < n_pairs)
            __builtin_nontemporal_store(c[v], o + 8 + v);
    }
}

extern "C" void kernel_launch(void* const* d_in, const int* in_sizes, int n_in,
                              void* d_out, int out_size, void* d_ws, size_t ws_size,
                              hipStream_t stream)
{
    const int*   loc = (const int*)  d_in[0];   // location (N_PAIRS, 2) int32
    const float* U   = (const float*)d_in[1];   // user_matrix (USER_NUM, 64)
    const float* G   = (const float*)d_in[2];   // goods_matrix (64, GOODS_NUM)
    float*       out = (float*)d_out;

    const int n_pairs = in_sizes[0] / 2;        // 1,000,000
    const int tiles   = (n_pairs + 15) / 16;    // 62,500
    const int blocks  = (tiles + 3) / 4;        // 4 waves (tiles) per 128-thr block

    hipLaunchKernelGGL(pair_dot_wmma_f32, dim3(blocks), dim3(128), 0, stream,
                       loc, U, G, out, n_pairs);
}